// PerFeatureEmbedding_31928786878606
// MI455X (gfx1250) — compile-verified
//
#include <hip/hip_runtime.h>
#include <hip/hip_bf16.h>

// y[bt, f, e] = x[bt, f] * W[f, e] + b[f, e]
// B*T = 32768, F = 64, E = 128, all f32. Pure store-bandwidth problem
// (1.07 GB out @ 23.3 TB/s => ~46us floor). We use V_WMMA_F32_16X16X4_F32 as a
// rank-1 FMA engine: A col0 = x[t0..t0+15, f], B row0 = W[f, e0..e0+15],
// C = bias tile; D's VGPR layout maps straight onto 8 coalesced row stores.

typedef __attribute__((ext_vector_type(2))) float v2f;
typedef __attribute__((ext_vector_type(8))) float v8f;

#define N_FEAT 64
#define EMBED  128

__global__ __launch_bounds__(256) void pfe_wmma_kernel(
    const float* __restrict__ x,      // [BT, 64]
    const float* __restrict__ W,      // [64, 128]
    const float* __restrict__ bias,   // [64, 128]
    float* __restrict__ out)          // [BT, 64, 128]
{
    const int lane = threadIdx.x & 31;
    const int wv   = threadIdx.x >> 5;   // wave id in block: 0..7 -> E tile
    const int e0   = wv << 4;            // 16-wide embed tile
    const int bt0  = blockIdx.x << 4;    // 16 rows of flattened (B*T)
    const int n    = lane & 15;          // N index within tile
    const int half = lane >> 4;          // 0: lanes 0-15, 1: lanes 16-31

    // A-matrix x gather: lanes 0-15 supply rows bt0..bt0+15 (K=0 column).
    // Lanes 16-31 read the same (valid) addresses; their contribution is
    // zeroed below with a branchless select so EXEC stays all-1s for WMMA.
    const float* xp = x + (size_t)(bt0 + n) * N_FEAT;

    // C/D layout: VGPR v <-> row M = v + 8*half, N = lane & 15.
    const size_t row0 = (size_t)(bt0 + half * 8);
    float* po = out + row0 * (size_t)(N_FEAT * EMBED) + e0 + n;

    for (int f = 0; f < N_FEAT; ++f) {
        // A: 16x4 f32. Lanes 0-15: VGPR0=K0, VGPR1=K1; lanes 16-31: K2,K3.
        // Only K=0 is live -> exact x*W in f32.
        float xv = xp[f];
        v2f a;
        a.x = half ? 0.0f : xv;   // v_cndmask, no EXEC divergence
        a.y = 0.0f;

        // B: 4x16 f32, row-striped across lanes (N = lane&15). Fill every
        // slot with W[f, e0+n]; rows K>=1 multiply against zero A columns,
        // so this is layout-robust.
        float wval = W[f * EMBED + e0 + n];
        v2f bm;
        bm.x = wval;
        bm.y = wval;

        // C: bias tile, constant along M for fixed (f, n).
        float cv = bias[f * EMBED + e0 + n];
        v8f c = {cv, cv, cv, cv, cv, cv, cv, cv};

        // D = A*B + C  (v_wmma_f32_16x16x4_f32)
        v8f d = __builtin_amdgcn_wmma_f32_16x16x4_f32(
            /*neg_a=*/false, a, /*neg_b=*/false, bm,
            /*c_mod=*/(short)0, c, /*reuse_a=*/false, /*reuse_b=*/false);

        // Store: D VGPR v holds y[bt0 + 8*half + v, f, e0+n].
        // 16 lanes x 4B = 64B contiguous per row per half-wave, streaming NT.
        float* q = po + (size_t)f * EMBED;
#pragma unroll
        for (int v = 0; v < 8; ++v)
            __builtin_nontemporal_store(d[v], q + (size_t)v * (N_FEAT * EMBED));
    }
}

extern "C" void kernel_launch(void* const* d_in, const int* in_sizes, int n_in,
                              void* d_out, int out_size, void* d_ws, size_t ws_size,
                              hipStream_t stream) {
    const float* x = (const float*)d_in[0];   // [16, 2048, 64]
    const float* W = (const float*)d_in[1];   // [64, 128]
    const float* b = (const float*)d_in[2];   // [64, 128]
    float* out = (float*)d_out;               // [16, 2048, 64, 128]

    const int BT = in_sizes[0] / N_FEAT;      // 32768
    dim3 grid(BT / 16);                       // 2048 blocks
    dim3 block(256);                          // 8 waves: one per 16-wide E tile
    hipLaunchKernelGGL(pfe_wmma_kernel, grid, block, 0, stream, x, W, b, out);
}